// LightGCN_Fusion_50362786513119
// MI455X (gfx1250) — compile-verified
//
#include <hip/hip_runtime.h>

typedef __attribute__((ext_vector_type(2))) float v2f;
typedef __attribute__((ext_vector_type(8))) float v8f;

#define NUM_USERS   100000
#define NUM_ITEMS   50000
#define NUM_BRANDS  1000
#define NTOT        151000            // NUM_USERS + NUM_ITEMS + NUM_BRANDS
#define E_EDGES     4800000
#define DIM         64
#define D_CONTENT   384
#define KDIM        448               // DIM + D_CONTENT

// ---------------------------------------------------------------------------
// Fusion GEMM: fused_item = leaky_relu(concat(item_id, content) @ W.T + b)
// One 128-thread block (4 waves) per 16-row M tile; wave w owns N-tile w.
// V_WMMA_F32_16X16X4_F32, K split at the item_id/content boundary (K=64) so
// the hot loop is exactly 2x global_load_b64 + 1x v_wmma, no pointer selects.
// ---------------------------------------------------------------------------
__global__ __launch_bounds__(128) void fuse_gemm_kernel(
    const float* __restrict__ item_id,     // [NUM_ITEMS, 64]
    const float* __restrict__ content,     // [NUM_ITEMS, 384]
    const float* __restrict__ W,           // [64, 448] row-major (we need W.T)
    const float* __restrict__ bias,        // [64]
    float* __restrict__ ego)               // [NTOT, 64]
{
    const int wave = threadIdx.x >> 5;
    const int lane = threadIdx.x & 31;
    const int half = lane >> 4;            // 0: lanes 0-15, 1: lanes 16-31
    const int l16  = lane & 15;

    const int m  = blockIdx.x * 16 + l16;  // A row held by this lane
    const int nb = wave * 16;              // N tile base for this wave
    const int n  = nb + l16;               // B/C column held by this lane

    v8f c = {0.f, 0.f, 0.f, 0.f, 0.f, 0.f, 0.f, 0.f};

    const float* wrow    = W + (size_t)n * KDIM;
    const float* arow_id = item_id + (size_t)m * DIM;
    const float* arow_ct = content + (size_t)m * D_CONTENT;

    // K in [0, 64): A from item_id_emb
    #pragma unroll 4
    for (int kb = 0; kb < DIM; kb += 4) {
        const int k = kb + 2 * half;
        v2f a = *(const v2f*)(arow_id + k);        // a.x=A[m][k], a.y=A[m][k+1]
        v2f b = *(const v2f*)(wrow + k);           // b.x=W[n][k], b.y=W[n][k+1]
        c = __builtin_amdgcn_wmma_f32_16x16x4_f32(
                false, a, false, b, (short)0, c, false, false);
    }
    // K in [64, 448): A from item_content_emb
    #pragma unroll 4
    for (int kb = DIM; kb < KDIM; kb += 4) {
        const int k = kb + 2 * half;
        v2f a = *(const v2f*)(arow_ct + (k - DIM));
        v2f b = *(const v2f*)(wrow + k);
        c = __builtin_amdgcn_wmma_f32_16x16x4_f32(
                false, a, false, b, (short)0, c, false, false);
    }

    const float bv = bias[n];
    float* out = ego + (size_t)NUM_USERS * DIM;
    #pragma unroll
    for (int r = 0; r < 8; ++r) {
        const int mr = blockIdx.x * 16 + r + 8 * half;   // C layout: VGPR r -> row r / 8+r
        float v = c[r] + bv;
        v = (v > 0.f) ? v : 0.01f * v;                   // leaky_relu, slope 0.01
        out[(size_t)mr * DIM + n] = v;
    }
}

// ---------------------------------------------------------------------------
// Assemble ego = concat(user, fused_item(already in ego), brand); acc = ego.
// float4 sweep (all region boundaries are multiples of 4 elements).
// ---------------------------------------------------------------------------
__global__ void assemble_kernel(const float4* __restrict__ user,
                                const float4* __restrict__ brand,
                                float4* __restrict__ ego,
                                float4* __restrict__ acc, int n4)
{
    const int i = blockIdx.x * blockDim.x + threadIdx.x;
    if (i >= n4) return;
    const int uN = NUM_USERS * DIM / 4;
    const int iN = (NUM_USERS + NUM_ITEMS) * DIM / 4;
    float4 v;
    if (i < uN)      v = user[i];
    else if (i < iN) v = ego[i];
    else             v = brand[i - iN];
    ego[i] = v;
    acc[i] = v;
}

__global__ void zero_kernel(float4* __restrict__ p, int n4)
{
    const int i = blockIdx.x * blockDim.x + threadIdx.x;
    if (i < n4) p[i] = make_float4(0.f, 0.f, 0.f, 0.f);
}

// ---------------------------------------------------------------------------
// SpMM layer: next[row] += val * cur[col].  One wave per edge (32 lanes x
// float2 = the full 64-dim row): row/col/val uniform per wave, coalesced
// 256B gather (the 38.7MB ego matrix is resident in the 192MB L2), scatter
// via guaranteed hardware global_atomic_add_f32 at device scope.
// ---------------------------------------------------------------------------
__device__ __forceinline__ void gatomic_fadd(float* addr, float val)
{
    asm volatile("global_atomic_add_f32 %0, %1, off scope:SCOPE_DEV"
                 :: "v"(addr), "v"(val) : "memory");
}

__global__ void spmm_kernel(const int* __restrict__ erow,
                            const int* __restrict__ ecol,
                            const float* __restrict__ eval,
                            const float* __restrict__ cur,
                            float* __restrict__ next)
{
    const long long t = (long long)blockIdx.x * blockDim.x + threadIdx.x;
    if (t >= (long long)E_EDGES * 32) return;
    const int e  = (int)(t >> 5);
    const int d2 = (int)(t & 31) * 2;

    const int   r    = erow[e];
    const int   cidx = ecol[e];
    const float v    = eval[e];

    const float2 x = *(const float2*)(cur + (size_t)cidx * DIM + d2);
    float* dst = next + (size_t)r * DIM + d2;
    gatomic_fadd(dst,     v * x.x);
    gatomic_fadd(dst + 1, v * x.y);
}

__global__ void axpy_kernel(float4* __restrict__ acc, const float4* __restrict__ x, int n4)
{
    const int i = blockIdx.x * blockDim.x + threadIdx.x;
    if (i >= n4) return;
    float4 a = acc[i];
    const float4 b = x[i];
    a.x += b.x; a.y += b.y; a.z += b.z; a.w += b.w;
    acc[i] = a;
}

// ---------------------------------------------------------------------------
// out[0:NTOT*64] = acc/4; then verbatim copies of user_emb and item_id_emb.
// ---------------------------------------------------------------------------
__global__ void finalize_kernel(float4* __restrict__ out,
                                const float4* __restrict__ user,
                                const float4* __restrict__ item_id, int n4)
{
    const int i = blockIdx.x * blockDim.x + threadIdx.x;
    if (i >= n4) return;
    const int b1 = NTOT * DIM / 4;                 // acc region
    const int b2 = b1 + NUM_USERS * DIM / 4;       // user_emb copy region
    if (i < b1) {
        float4 a = out[i];
        a.x *= 0.25f; a.y *= 0.25f; a.z *= 0.25f; a.w *= 0.25f;
        out[i] = a;
    } else if (i < b2) {
        out[i] = user[i - b1];
    } else {
        out[i] = item_id[i - b2];
    }
}

extern "C" void kernel_launch(void* const* d_in, const int* in_sizes, int n_in,
                              void* d_out, int out_size, void* d_ws, size_t ws_size,
                              hipStream_t stream)
{
    const int*   erow    = (const int*)  d_in[0];
    const int*   ecol    = (const int*)  d_in[1];
    const float* eval    = (const float*)d_in[2];
    const float* user    = (const float*)d_in[3];
    const float* item_id = (const float*)d_in[4];
    const float* brand   = (const float*)d_in[5];
    const float* content = (const float*)d_in[6];
    const float* W       = (const float*)d_in[7];
    const float* bias    = (const float*)d_in[8];

    float* acc = (float*)d_out;                       // first NTOT*64 of d_out is acc

    float* ego_a = (float*)d_ws;                      // ping-pong ego buffers
    float* ego_b = ego_a + (size_t)NTOT * DIM;

    const int n1 = NTOT * DIM;
    const int n4 = n1 / 4;

    // 1) Fusion GEMM (WMMA f32) -> item region of ego_a
    fuse_gemm_kernel<<<NUM_ITEMS / 16, 128, 0, stream>>>(item_id, content, W, bias, ego_a);

    // 2) ego_a = concat(user, fused, brand); acc = ego_a
    assemble_kernel<<<(n4 + 255) / 256, 256, 0, stream>>>(
        (const float4*)user, (const float4*)brand, (float4*)ego_a, (float4*)acc, n4);

    // 3) 3 propagation layers
    float* cur = ego_a;
    float* nxt = ego_b;
    const long long spmm_threads = (long long)E_EDGES * 32;
    const int spmm_blocks = (int)((spmm_threads + 255) / 256);
    for (int layer = 0; layer < 3; ++layer) {
        zero_kernel<<<(n4 + 255) / 256, 256, 0, stream>>>((float4*)nxt, n4);
        spmm_kernel<<<spmm_blocks, 256, 0, stream>>>(erow, ecol, eval, cur, nxt);
        axpy_kernel<<<(n4 + 255) / 256, 256, 0, stream>>>((float4*)acc, (const float4*)nxt, n4);
        float* tmp = cur; cur = nxt; nxt = tmp;
    }

    // 4) scale by 1/4 and append the verbatim embedding copies
    const int out4 = out_size / 4;
    finalize_kernel<<<(out4 + 255) / 256, 256, 0, stream>>>(
        (float4*)acc, (const float4*)user, (const float4*)item_id, out4);
}